// GCN_CAM_81088982548931
// MI455X (gfx1250) — compile-verified
//
#include <hip/hip_runtime.h>
#include <hip/hip_bf16.h>

typedef __attribute__((ext_vector_type(2))) float v2f;
typedef __attribute__((ext_vector_type(8))) float v8f;

#define GCN_N   102400
#define GCN_E   1638400
#define NODES_PER_GRAPH 400
#define BN_EPS  1e-5f

// ---------------------------------------------------------------------------
// Degree / normalization: dinv[i] = rsqrt(1 + #edges with dst==i)
// ---------------------------------------------------------------------------
__global__ void deg_init(float* deg, int n) {
    int i = blockIdx.x * blockDim.x + threadIdx.x;
    if (i < n) deg[i] = 1.0f;                       // self-loop
}
__global__ void deg_accum(const int* __restrict__ dst, float* deg, int e) {
    int i = blockIdx.x * blockDim.x + threadIdx.x;
    if (i < e) atomicAdd(&deg[dst[i]], 1.0f);
}
__global__ void deg_finish(float* deg, int n) {
    int i = blockIdx.x * blockDim.x + threadIdx.x;
    if (i < n) deg[i] = rsqrtf(deg[i]);
}

// ---------------------------------------------------------------------------
// GEMM: M_[N x DO] = H[N x K] @ W[K x DO], fp32 WMMA 16x16x4.
// One wave -> 16 rows x full DO. A read once from HBM; W staged in LDS.
// Layouts per CDNA5 ISA 7.12.2 (wave32):
//   A 16x4 f32 : lane m=lane&15, half=lane>>4; vgpr v holds K = k0 + 2*half + v
//   B 4x16 f32 : lane n=lane&15, half=lane>>4; vgpr v holds K = k0 + 2*half + v
//   C/D 16x16  : lane n=lane&15; vgpr r holds M = r + 8*(lane>>4)
// ---------------------------------------------------------------------------
template <int DO>
__global__ __launch_bounds__(256) void gcn_gemm(const float* __restrict__ H,
                                                const float* __restrict__ W,
                                                float* __restrict__ M_,
                                                int K) {
    extern __shared__ float ldsW[];                 // K*DO floats
    const int tid = threadIdx.x;
    for (int i = tid; i < K * DO; i += 256) ldsW[i] = W[i];
    __syncthreads();

    const int wave = tid >> 5;
    const int lane = tid & 31;
    const int m    = lane & 15;
    const int half = lane >> 4;                     // 0/1
    const int rowTile = blockIdx.x * 8 + wave;      // 16-row tile
    const float* Arow = H + (size_t)(rowTile * 16 + m) * K;

    v8f acc[DO / 16] = {};

    for (int k0 = 0; k0 < K; k0 += 4) {
        v2f a;
        a.x = Arow[k0 + 2 * half + 0];
        a.y = Arow[k0 + 2 * half + 1];
#pragma unroll
        for (int t = 0; t < DO / 16; ++t) {
            const int n = t * 16 + m;               // B column = lane&15
            v2f b;
            b.x = ldsW[(k0 + 2 * half + 0) * DO + n];
            b.y = ldsW[(k0 + 2 * half + 1) * DO + n];
            acc[t] = __builtin_amdgcn_wmma_f32_16x16x4_f32(
                false, a, false, b, (short)0, acc[t], false, false);
        }
    }

    float* Cbase = M_ + (size_t)(rowTile * 16) * DO;
#pragma unroll
    for (int t = 0; t < DO / 16; ++t) {
#pragma unroll
        for (int r = 0; r < 8; ++r) {
            const int mo = r + 8 * half;
            Cbase[(size_t)mo * DO + t * 16 + m] = acc[t][r];
        }
    }
}

// ---------------------------------------------------------------------------
// agg init with fused self-loop: agg[i][c] = m[i][c] * dinv[i]^2
// ---------------------------------------------------------------------------
template <int DO>
__global__ void agg_init(const float* __restrict__ m, const float* __restrict__ dinv,
                         float* __restrict__ agg, int n) {
    int idx = blockIdx.x * blockDim.x + threadIdx.x;
    if (idx >= n * DO) return;
    const int i = idx / DO;
    const float w = dinv[i];
    agg[idx] = m[idx] * w * w;
}

// ---------------------------------------------------------------------------
// Edge scatter: agg[dst] += m[src] * dinv[src]*dinv[dst]
// Thread = (edge, 4-channel group); float4 gather + 4 f32 atomics.
// ---------------------------------------------------------------------------
template <int DO>
__global__ void scatter_edges(const int* __restrict__ src, const int* __restrict__ dst,
                              const float* __restrict__ dinv, const float* __restrict__ m,
                              float* __restrict__ agg, int e) {
    const int G = DO / 4;
    long idx = (long)blockIdx.x * blockDim.x + threadIdx.x;
    if (idx >= (long)e * G) return;
    const int ei = (int)(idx / G);
    const int g  = (int)(idx % G);
    const int s = src[ei];
    const int d = dst[ei];
    const float w = dinv[s] * dinv[d];
    const float4 v = *((const float4*)(m + (size_t)s * DO) + g);
    float* ad = agg + (size_t)d * DO + g * 4;
    atomicAdd(ad + 0, v.x * w);
    atomicAdd(ad + 1, v.y * w);
    atomicAdd(ad + 2, v.z * w);
    atomicAdd(ad + 3, v.w * w);
}

// ---------------------------------------------------------------------------
// h = bn(agg + b) [+ relu]   (in place over agg)
// ---------------------------------------------------------------------------
template <bool RELU>
__global__ void bn_relu(float* __restrict__ h, const float* __restrict__ b,
                        const float* __restrict__ gamma, const float* __restrict__ beta,
                        const float* __restrict__ mean, const float* __restrict__ var,
                        int n, int DO) {
    int idx = blockIdx.x * blockDim.x + threadIdx.x;
    if (idx >= n * DO) return;
    const int c = idx - (idx / DO) * DO;
    const float s = gamma[c] * rsqrtf(var[c] + BN_EPS);
    float v = (h[idx] + b[c] - mean[c]) * s + beta[c];
    if (RELU) v = fmaxf(v, 0.0f);
    h[idx] = v;
}

// ---------------------------------------------------------------------------
// Mean pool over 400 nodes/graph then 128->2 linear. Block per graph.
// ---------------------------------------------------------------------------
__global__ __launch_bounds__(128) void pool_linear(const float* __restrict__ h,
                                                   const float* __restrict__ W,
                                                   const float* __restrict__ b,
                                                   float* __restrict__ out) {
    const int g = blockIdx.x;
    const int c = threadIdx.x;                       // channel 0..127
    const float* hg = h + (size_t)g * NODES_PER_GRAPH * 128 + c;
    float s = 0.0f;
    for (int nd = 0; nd < NODES_PER_GRAPH; ++nd) s += hg[(size_t)nd * 128];
    s *= (1.0f / NODES_PER_GRAPH);
    __shared__ float r0[128];
    __shared__ float r1[128];
    r0[c] = s * W[c * 2 + 0];
    r1[c] = s * W[c * 2 + 1];
    __syncthreads();
    for (int off = 64; off > 0; off >>= 1) {
        if (c < off) { r0[c] += r0[c + off]; r1[c] += r1[c + off]; }
        __syncthreads();
    }
    if (c == 0) {
        out[g * 2 + 0] = r0[0] + b[0];
        out[g * 2 + 1] = r1[0] + b[1];
    }
}

// ---------------------------------------------------------------------------
// Host-side per-layer driver
// ---------------------------------------------------------------------------
template <int DO>
static void run_layer(const float* hin, int K, const float* W, const float* bias,
                      const float* gamma, const float* beta, const float* mean,
                      const float* var, const int* src, const int* dst,
                      const float* dinv, float* mbuf, float* agg, bool relu,
                      hipStream_t stream) {
    gcn_gemm<DO><<<GCN_N / 16 / 8, 256, (size_t)K * DO * sizeof(float), stream>>>(hin, W, mbuf, K);
    int elems = GCN_N * DO;
    agg_init<DO><<<(elems + 255) / 256, 256, 0, stream>>>(mbuf, dinv, agg, GCN_N);
    long tot = (long)GCN_E * (DO / 4);
    scatter_edges<DO><<<(unsigned)((tot + 255) / 256), 256, 0, stream>>>(src, dst, dinv, mbuf, agg, GCN_E);
    if (relu)
        bn_relu<true><<<(elems + 255) / 256, 256, 0, stream>>>(agg, bias, gamma, beta, mean, var, GCN_N, DO);
    else
        bn_relu<false><<<(elems + 255) / 256, 256, 0, stream>>>(agg, bias, gamma, beta, mean, var, GCN_N, DO);
}

extern "C" void kernel_launch(void* const* d_in, const int* in_sizes, int n_in,
                              void* d_out, int out_size, void* d_ws, size_t ws_size,
                              hipStream_t stream) {
    const float* x  = (const float*)d_in[0];
    const int*   ei = (const int*)d_in[1];
    const int* src = ei;                 // edge_index[0]
    const int* dst = ei + GCN_E;         // edge_index[1]

    const float *W[5], *bias[5], *gamma[5], *beta[5], *mean[5], *var[5];
    for (int i = 0; i < 5; ++i) {
        W[i]     = (const float*)d_in[2 + 6 * i + 0];
        bias[i]  = (const float*)d_in[2 + 6 * i + 1];
        gamma[i] = (const float*)d_in[2 + 6 * i + 2];
        beta[i]  = (const float*)d_in[2 + 6 * i + 3];
        mean[i]  = (const float*)d_in[2 + 6 * i + 4];
        var[i]   = (const float*)d_in[2 + 6 * i + 5];
    }
    const float* linW = (const float*)d_in[32];
    const float* linb = (const float*)d_in[33];

    char* ws = (char*)d_ws;
    float* dinv = (float*)ws;  ws += (size_t)GCN_N * sizeof(float);
    float* mbuf = (float*)ws;  ws += (size_t)GCN_N * 128 * sizeof(float);
    float* bufA = (float*)ws;  ws += (size_t)GCN_N * 128 * sizeof(float);
    float* bufB = (float*)ws;

    // degrees -> dinv
    deg_init  <<<(GCN_N + 255) / 256, 256, 0, stream>>>(dinv, GCN_N);
    deg_accum <<<(GCN_E + 255) / 256, 256, 0, stream>>>(dst, dinv, GCN_E);
    deg_finish<<<(GCN_N + 255) / 256, 256, 0, stream>>>(dinv, GCN_N);

    // layer 0: 400 -> 32 (x -> bufA)
    run_layer<32>(x,    400, W[0], bias[0], gamma[0], beta[0], mean[0], var[0],
                  src, dst, dinv, mbuf, bufA, true, stream);
    // layer 1: 32 -> 32 (bufA -> bufB)
    run_layer<32>(bufA, 32,  W[1], bias[1], gamma[1], beta[1], mean[1], var[1],
                  src, dst, dinv, mbuf, bufB, true, stream);
    // layer 2: 32 -> 64 (bufB -> bufA)
    run_layer<64>(bufB, 32,  W[2], bias[2], gamma[2], beta[2], mean[2], var[2],
                  src, dst, dinv, mbuf, bufA, true, stream);
    // layer 3: 64 -> 64 (bufA -> bufB)
    run_layer<64>(bufA, 64,  W[3], bias[3], gamma[3], beta[3], mean[3], var[3],
                  src, dst, dinv, mbuf, bufB, true, stream);
    // layer 4: 64 -> 128 (bufB -> bufA), no relu
    run_layer<128>(bufB, 64, W[4], bias[4], gamma[4], beta[4], mean[4], var[4],
                   src, dst, dinv, mbuf, bufA, false, stream);

    // pool + linear -> logits [256 x 2]
    pool_linear<<<GCN_N / NODES_PER_GRAPH, 128, 0, stream>>>(bufA, linW, linb, (float*)d_out);
}